// FluidConstraints_2559800508972
// MI455X (gfx1250) — compile-verified
//
#include <hip/hip_runtime.h>
#include <hip/hip_bf16.h>

// SPH density via distance-matrix WMMA:
//   r^2(m,n) = |x_m|^2 + |x_n|^2 - 2 x_m . x_n
// maps onto V_WMMA_F32_16X16X4_F32 with
//   A row m = (-2x, -2y, -2z, 1), B col n = (x, y, z, |x_n|^2), C[m][n] = |x_m|^2.
//
// Branchless spline + cutoff:  with q = r/H,
//   w(q) = max( 2*(1-q)^3 + min(2q-1, 0)^3 , 0 )
// reproduces (q<=0.5: 1-6q^2+6q^3 ; q<1: 2(1-q)^3 ; q>=1: 0) exactly.
// Invalid columns are excluded by poisoning their |x_n|^2 WMMA slot to 1e8
// (-> r^2 huge -> w clamps to 0); invalid rows are zeroed after the loop.

typedef __attribute__((ext_vector_type(2))) float v2f;
typedef __attribute__((ext_vector_type(8))) float v8f;

#define TILE            16
#define WAVES_PER_BLOCK 8
#define THREADS         (WAVES_PER_BLOCK * 32)

// H = 0.1 ; sigma = 8/(pi*H^3)
#define INV_H   10.0f
#define SIGMA   2546.4790894703255f

__global__ __launch_bounds__(THREADS)
void sph_density_wmma(const float* __restrict__ locs,
                      const int*  __restrict__ npc,
                      float*      __restrict__ out,
                      int N) {
    const int nTiles = N / TILE;
    const int b      = blockIdx.x / nTiles;
    const int mt     = blockIdx.x % nTiles;
    const int m0     = mt * TILE;

    const int tid    = threadIdx.x;
    const int lane   = tid & 31;
    const int wave   = tid >> 5;
    const int half   = lane >> 4;      // 0: lanes 0-15, 1: lanes 16-31
    const int l16    = lane & 15;
    const int rbase  = half * 8;       // row offset of this half's C/D registers

    const int nvalid = npc[b];
    const float* base = locs + (size_t)b * N * 3;

    __shared__ float rowNorm[TILE];
    __shared__ float rowSum[TILE];
    if (tid < TILE) {
        const float* pm = base + (size_t)(m0 + tid) * 3;
        float x = pm[0], y = pm[1], z = pm[2];
        rowNorm[tid] = x * x + y * y + z * z;
        rowSum[tid]  = 0.0f;
    }
    __syncthreads();

    // ---- A fragment (16x4 f32): lane l holds row m0+(l&15); K={0,1} in lanes
    // 0-15, K={2,3} in lanes 16-31. Encodes (-2x, -2y, -2z, 1).
    v2f afrag;
    {
        const float* pm = base + (size_t)(m0 + l16) * 3;
        float x = pm[0], y = pm[1], z = pm[2];
        afrag.x = half ? (-2.0f * z) : (-2.0f * x);
        afrag.y = half ? 1.0f        : (-2.0f * y);
    }

    // ---- C fragment: C[m][n] = |x_m|^2 (row-broadcast), 8 regs per lane.
    v8f cfrag;
#pragma unroll
    for (int r = 0; r < 8; ++r) cfrag[r] = rowNorm[rbase + r];

    float acc[8];
#pragma unroll
    for (int r = 0; r < 8; ++r) acc[r] = 0.0f;

    // ---- Sweep column tiles: wave w handles tiles w, w+8, w+16, ...
    for (int jt = wave; jt < nTiles; jt += WAVES_PER_BLOCK) {
        const int p = jt * TILE + l16;
        const float* pp = base + (size_t)p * 3;
        float x = pp[0], y = pp[1], z = pp[2];

        const bool vj = (p < nvalid);
        float nrm = fmaf(x, x, fmaf(y, y, z * z));

        // B fragment (4x16 f32): col n=p; K={0,1} lanes 0-15, K={2,3} lanes
        // 16-31. Encodes (x, y, z, |x_n|^2); invalid col -> |x_n|^2 = 1e8.
        v2f bfrag;
        bfrag.x = half ? z : x;
        bfrag.y = half ? (vj ? nrm : 1.0e8f) : y;

        // D[m][n] = A.B + C = r^2(m,n) for the whole 16x16 tile.
        v8f d = __builtin_amdgcn_wmma_f32_16x16x4_f32(
            /*neg_a=*/false, afrag, /*neg_b=*/false, bfrag,
            /*c_mod=*/(short)0, cfrag, /*reuse_a=*/false, /*reuse_b=*/false);

#pragma unroll
        for (int r = 0; r < 8; ++r) {
            float r2 = fmaxf(d[r], 0.0f);                 // clamp FP noise
            float q  = __builtin_amdgcn_sqrtf(r2) * INV_H; // raw v_sqrt_f32
            float t  = 1.0f - q;
            float s  = fminf(fmaf(2.0f, q, -1.0f), 0.0f);
            float t3 = t * t * t;
            float s3 = s * s * s;
            acc[r] += fmaxf(fmaf(2.0f, t3, s3), 0.0f);    // spline + (r<H) mask
        }
    }

    // ---- Row validity (loop-invariant): zero out invalid rows once.
#pragma unroll
    for (int r = 0; r < 8; ++r)
        acc[r] = ((m0 + rbase + r) < nvalid) ? acc[r] : 0.0f;

    // ---- Row-sum reduction: columns of a row live across the 16 lanes of one
    // half-wave (rows m0..m0+7 in lanes 0-15, rows m0+8..m0+15 in lanes 16-31).
#pragma unroll
    for (int r = 0; r < 8; ++r) {
        acc[r] += __shfl_xor(acc[r], 8, 32);
        acc[r] += __shfl_xor(acc[r], 4, 32);
        acc[r] += __shfl_xor(acc[r], 2, 32);
        acc[r] += __shfl_xor(acc[r], 1, 32);
    }

    if (l16 == 0) {  // lane 0 and lane 16 of each wave carry the half-sums
#pragma unroll
        for (int r = 0; r < 8; ++r)
            atomicAdd(&rowSum[rbase + r], acc[r]);  // ds_add_f32
    }
    __syncthreads();

    if (tid < TILE) {
        out[(size_t)b * N + m0 + tid] = rowSum[tid] * SIGMA;
    }
}

extern "C" void kernel_launch(void* const* d_in, const int* in_sizes, int n_in,
                              void* d_out, int out_size, void* d_ws, size_t ws_size,
                              hipStream_t stream) {
    const float* locs = (const float*)d_in[0];       // [B, N, 3] f32
    const int*   npc  = (const int*)d_in[1];         // [B] i32
    float*       out  = (float*)d_out;               // [B, N] f32

    const int B = in_sizes[1];
    const int N = in_sizes[0] / (3 * B);             // 4096
    const int nTiles = N / TILE;                     // 256

    dim3 grid(B * nTiles);
    dim3 block(THREADS);
    hipLaunchKernelGGL(sph_density_wmma, grid, block, 0, stream,
                       locs, npc, out, N);
}